// GraphConv2d_57458072486033
// MI455X (gfx1250) — compile-verified
//
#include <hip/hip_runtime.h>
#include <hip/hip_bf16.h>

// Problem constants (match reference)
#define BDIM 4
#define CDIM 64
#define NDIM 8192
#define KNBR 16
#define COUTC 64
#define NLOG2 13          // N = 8192
#define EPSV 1e-5f
#define NEG_SLOPE 0.2f

typedef __attribute__((ext_vector_type(2))) float v2f;
typedef __attribute__((ext_vector_type(4))) float v4f;
typedef __attribute__((ext_vector_type(8))) float v8f;

// ---------------------------------------------------------------------------
// Stage 1: y1[m][o] = sum_c (W[o][c]-W[o][c+64]) * x[b][c][n]    (o in 0..63)
//          y2[m][o] = sum_c  W[o][c+64]          * x[b][c][n]
// m = b*N + n flat column. V_WMMA_F32_16X16X4_F32, one wave per 16x16 tile,
// 8 waves/block cover combined rows o' = 0..127 for one m-tile.
//
// Weight fragments are precomputed branchlessly before the K-loop (both W
// halves are always loaded, v_cndmask selects), so the steady-state loop is
// just: 2 coalesced x loads -> wait -> wmma.
// ---------------------------------------------------------------------------
__global__ __launch_bounds__(256)
void edgeconv_pretransform_wmma(const float* __restrict__ x,
                                const float* __restrict__ W,
                                float* __restrict__ y1,
                                float* __restrict__ y2)
{
    const int lane  = threadIdx.x & 31;
    const int wave  = threadIdx.x >> 5;     // 0..7
    const int o0    = wave * 16;            // combined-row tile base (0..112)
    const int m0    = blockIdx.x * 16;      // flat column base
    const int b     = m0 >> NLOG2;
    const int n0    = m0 & (NDIM - 1);

    const int mrow  = lane & 15;            // A: M row held by this lane
    const int khalf = (lane >> 4) * 2;      // A/B: K sub-offset (0 or 2)
    const int ncol  = lane & 15;            // B: N column held by this lane

    const int  op      = o0 + mrow;         // combined output row 0..127
    const int  r       = op & 63;           // physical W row
    const bool lowhalf = (op < 64);

    // ---- Precompute all 16 A-fragments, branchless --------------------------
    v2f afrag[16];
    #pragma unroll
    for (int ks = 0; ks < 16; ++ks) {
        #pragma unroll
        for (int v = 0; v < 2; ++v) {
            const int  c  = ks * 4 + khalf + v;        // input channel 0..63
            const float wa = W[r * 128 + c];           // W1[r][c]
            const float wb = W[r * 128 + 64 + c];      // W2[r][c]
            afrag[ks][v] = lowhalf ? (wa - wb) : wb;   // v_cndmask, no branch
        }
    }

    // ---- K-loop: load B tile fragment, accumulate with WMMA ----------------
    const float* xb = x + (size_t)b * CDIM * NDIM + n0 + ncol;

    v8f acc = {};
    #pragma unroll
    for (int ks = 0; ks < 16; ++ks) {
        v2f bv;
        const int c = ks * 4 + khalf;
        bv[0] = xb[(size_t)(c    ) * NDIM];
        bv[1] = xb[(size_t)(c + 1) * NDIM];
        acc = __builtin_amdgcn_wmma_f32_16x16x4_f32(
            /*neg_a=*/false, afrag[ks], /*neg_b=*/false, bv,
            /*c_mod=*/(short)0, acc, /*reuse_a=*/false, /*reuse_b=*/false);
    }

    // C/D layout: lane<16 holds M=v (o'=o0+v), lane>=16 holds M=v+8,
    // column m = m0 + (lane & 15). Accumulators are contiguous in o in the
    // [m][64] layout -> two 16B stores per lane.
    const int   mcol = m0 + (lane & 15);
    const int   ohi  = (lane >= 16) ? 8 : 0;
    float*      dst  = (o0 < 64) ? y1 : y2;
    const int   och  = ((o0 < 64) ? o0 : (o0 - 64)) + ohi;
    float*      p    = dst + (size_t)mcol * COUTC + och;
    v4f lo = { acc[0], acc[1], acc[2], acc[3] };
    v4f hi = { acc[4], acc[5], acc[6], acc[7] };
    *(v4f*)(p)     = lo;
    *(v4f*)(p + 4) = hi;
}

// ---------------------------------------------------------------------------
// Stage 2: per node m = b*N+n, per channel o:
//   out[b][o][n] = max_k leaky( (y1[b,i(k)] + y2[b,j(k)])[o] * inv[o] + bias[o] )
// 64 threads per node (one per channel) -> coalesced 256B gathers from the
// L2-resident y tables. Edge indices staged once through LDS.
// ---------------------------------------------------------------------------
__global__ __launch_bounds__(256)
void edgeconv_aggregate(const long long* __restrict__ eidx,
                        const float* __restrict__ y1,
                        const float* __restrict__ y2,
                        const float* __restrict__ gamma,
                        const float* __restrict__ beta,
                        const float* __restrict__ rmean,
                        const float* __restrict__ rvar,
                        float* __restrict__ out)
{
    __shared__ int s_i[4][KNBR];   // edge_index[1] -> i (center)
    __shared__ int s_j[4][KNBR];   // edge_index[0] -> j (neighbor)

    const int local = threadIdx.x & 63;     // channel o
    const int slot  = threadIdx.x >> 6;     // node slot within block (0..3)
    const int m     = blockIdx.x * 4 + slot;

    const size_t BNK = (size_t)BDIM * NDIM * KNBR;
    if (local < KNBR) {
        s_j[slot][local] = (int)eidx[(size_t)m * KNBR + local];
        s_i[slot][local] = (int)eidx[BNK + (size_t)m * KNBR + local];
    }
    __syncthreads();

    const int o = local;
    const int b = m >> NLOG2;
    const int n = m & (NDIM - 1);

    const float inv  = gamma[o] * rsqrtf(rvar[o] + EPSV);
    const float bias = beta[o] - rmean[o] * inv;

    const float* yb1 = y1 + (size_t)b * NDIM * COUTC;
    const float* yb2 = y2 + (size_t)b * NDIM * COUTC;

    float best = -3.402823466e+38f;
    #pragma unroll
    for (int k = 0; k < KNBR; ++k) {
        const int ii = s_i[slot][k];
        const int jj = s_j[slot][k];
        float v = yb1[(size_t)ii * COUTC + o] + yb2[(size_t)jj * COUTC + o];
        float h = fmaf(v, inv, bias);
        h = (h >= 0.0f) ? h : NEG_SLOPE * h;
        best = fmaxf(best, h);
    }
    out[(size_t)b * COUTC * NDIM + (size_t)o * NDIM + n] = best;
}

// ---------------------------------------------------------------------------
extern "C" void kernel_launch(void* const* d_in, const int* in_sizes, int n_in,
                              void* d_out, int out_size, void* d_ws, size_t ws_size,
                              hipStream_t stream) {
    const float*     x     = (const float*)d_in[0];
    const long long* eidx  = (const long long*)d_in[1];   // int64 indices
    const float*     W     = (const float*)d_in[2];
    const float*     gamma = (const float*)d_in[3];
    const float*     beta  = (const float*)d_in[4];
    const float*     rmean = (const float*)d_in[5];
    const float*     rvar  = (const float*)d_in[6];

    float* y1 = (float*)d_ws;                               // [B*N][64], 8 MB
    float* y2 = y1 + (size_t)BDIM * NDIM * COUTC;           // [B*N][64], 8 MB

    const int m_total = BDIM * NDIM;                        // 32768 columns

    edgeconv_pretransform_wmma<<<m_total / 16, 256, 0, stream>>>(x, W, y1, y2);
    edgeconv_aggregate<<<m_total / 4, 256, 0, stream>>>(eidx, y1, y2, gamma, beta,
                                                        rmean, rvar, (float*)d_out);
}